// Pooling_2D_density_3D_841813590233
// MI455X (gfx1250) — compile-verified
//
#include <hip/hip_runtime.h>
#include <stdint.h>

// Problem constants from the reference (I=16, O=8, J=4, B=16)
#define BATCH        16
#define N_IN         1024   // 16*16*4
#define N_OUT        256    // 8*8*4
#define K_PROJ       81     // 1 + 2*8 + 64
#define ROWS_PER_BLK 16

typedef unsigned int u32;
typedef u32 __attribute__((ext_vector_type(4))) v4u;
typedef int __attribute__((ext_vector_type(8))) v8i;
typedef int __attribute__((ext_vector_type(4))) v4i;

// out[b,r,c] = sum_k [cols[k,r]>=0 && cols[k,c]>=0] * X[b, cols[k,r], cols[k,c]]
// (mask[k,*] == (cols[k,*] >= 0) by construction of the projector table)
__global__ __launch_bounds__(256) void pool2d_proj3d_kernel(
    const float* __restrict__ X,      // (B, N_IN, N_IN) f32
    const int*   __restrict__ cols,   // (K_PROJ, N_OUT) i32, -1 = masked
    float*       __restrict__ out)    // (B, N_OUT, N_OUT) f32
{
    __shared__ int colr[K_PROJ * ROWS_PER_BLK];  // cols[k, r0+j] tile, 5184 B

    const int b  = blockIdx.x >> 4;                 // 16 row-tiles per batch
    const int r0 = (blockIdx.x & 15) * ROWS_PER_BLK;
    const int c  = threadIdx.x;                     // output column

    // ---- TDM: DMA cols[0..80, r0..r0+15] (2-D strided tile) into LDS ----
    // D# per CDNA5 ISA 8.3/8.4: data_size=4B, tile_dim0=16, tile_dim1=81,
    // tensor_dim0_stride=256; tensor_dims set large (tile is always in-bounds).
    if (threadIdx.x < 32) {  // one wave issues (redundant issue would also be correct)
        uint64_t gaddr = (uint64_t)(uintptr_t)(cols + r0);
        u32 ldsa = (u32)(uintptr_t)(&colr[0]);      // flat LDS addr: low 32 bits = LDS byte offset
        v4u g0;
        g0.x = 1u;                                   // count = 1 valid descriptor
        g0.y = ldsa;                                 // lds_addr
        g0.z = (u32)gaddr;                           // global_addr[31:0]
        g0.w = (u32)(gaddr >> 32) | (2u << 30);      // global_addr[56:32] | type=2 ("image")
        v8i g1;
        g1.s0 = (int)(2u << 16);                     // workgroup_mask=0, data_size=2 (4 bytes)
        g1.s1 = 0;                                   // no atomic barrier; tensor_dim0[15:0]=0
        g1.s2 = 16;                                  // tensor_dim0 = 1<<20 (hi16=16); tensor_dim1 lo16=0
        g1.s3 = (int)(0x10u | (16u << 16));          // tensor_dim1 hi16=16 (=1<<20); tile_dim0=16
        g1.s4 = K_PROJ;                              // tile_dim1 = 81; tile_dim2 = 0 (2-D)
        g1.s5 = N_OUT;                               // tensor_dim0_stride = 256 elements
        g1.s6 = 0;                                   // stride hi / tensor_dim1_stride (unused, 2-D)
        g1.s7 = 0;
        v4i g2 = {0, 0, 0, 0};                       // groups 2/3 unused for 2-D tensor
        v4i g3 = {0, 0, 0, 0};
        v8i g4 = {0, 0, 0, 0, 0, 0, 0, 0};           // extra group (clang-23 6-arg form), disabled
        __builtin_amdgcn_tensor_load_to_lds(g0, g1, g2, g3, g4, 0);
        __builtin_amdgcn_s_wait_tensorcnt(0);
    }
    __syncthreads();

    float acc[ROWS_PER_BLK];
#pragma unroll
    for (int j = 0; j < ROWS_PER_BLK; ++j) acc[j] = 0.0f;

    const float* __restrict__ Xb = X + (size_t)b * N_IN * N_IN;

    for (int k = 0; k < K_PROJ; ++k) {
        const int cc = cols[k * N_OUT + c];          // L2-broadcast across blocks
        if (cc < 0) continue;                        // kills ~77/81 iterations per lane
        const float* __restrict__ Xcol = Xb + cc;
#pragma unroll
        for (int j = 0; j < ROWS_PER_BLK; ++j) {
            const int cr = colr[k * ROWS_PER_BLK + j];   // uniform ds_load broadcast
            if (cr >= 0) acc[j] += Xcol[(size_t)cr * N_IN];
        }
    }

    float* __restrict__ ob = out + (size_t)b * N_OUT * N_OUT + (size_t)r0 * N_OUT + c;
#pragma unroll
    for (int j = 0; j < ROWS_PER_BLK; ++j) ob[j * N_OUT] = acc[j];   // coalesced
}

extern "C" void kernel_launch(void* const* d_in, const int* in_sizes, int n_in,
                              void* d_out, int out_size, void* d_ws, size_t ws_size,
                              hipStream_t stream) {
    (void)in_sizes; (void)n_in; (void)out_size; (void)d_ws; (void)ws_size;
    const float* X    = (const float*)d_in[0];   // (16, 1024, 1024) f32
    const int*   cols = (const int*)d_in[1];     // (81, 256) i32 (-1 where masked)
    // d_in[2] (mask) is redundant: mask == (cols >= 0)
    float* out = (float*)d_out;                  // (16, 256, 256) f32

    dim3 grid(BATCH * (N_OUT / ROWS_PER_BLK));   // 256 blocks
    dim3 block(256);                             // 8 wave32s
    pool2d_proj3d_kernel<<<grid, block, 0, stream>>>(X, cols, out);
}